// NaSingleOp_3959959847491
// MI455X (gfx1250) — compile-verified
//
#include <hip/hip_runtime.h>

#define N_NODES 100000
#define N_EDGES 600000
#define DIM 128
#define WSTRIDE 136   // padded LDS row stride (floats) for weight matrices
#define TSTRIDE 132   // padded LDS row stride (floats) for staging tiles
#define NSLABS (N_NODES / 16)        // 6250, exact
#define WAVES_PER_BLOCK 8

typedef __attribute__((ext_vector_type(2))) float v2f;
typedef __attribute__((ext_vector_type(8))) float v8f;

// ---------------------------------------------------------------------------
// Kernel 1: h = (1+eps)*x with eps=0  ->  h = x   (vectorized copy into d_ws)
// ---------------------------------------------------------------------------
__global__ void init_h_kernel(const float* __restrict__ x,
                              float* __restrict__ h, int n4) {
  int i = blockIdx.x * blockDim.x + threadIdx.x;
  if (i < n4) ((float4*)h)[i] = ((const float4*)x)[i];
}

// ---------------------------------------------------------------------------
// Kernel 2: edge scatter-add. One wave per edge: 32 lanes x float4 = 128 cols.
// Coalesced 512B gather of x[src], 4 global_atomic_add_f32 per lane into h[dst].
// ---------------------------------------------------------------------------
__global__ void scatter_kernel(const float* __restrict__ x,
                               const long long* __restrict__ ei, // [2, E]
                               const float* __restrict__ ew,
                               float* __restrict__ h) {
  int wave = (blockIdx.x * blockDim.x + threadIdx.x) >> 5;
  if (wave >= N_EDGES) return;
  int lane = threadIdx.x & 31;
  long long s = ei[wave];            // src
  long long d = ei[N_EDGES + wave];  // dst
  float w = ew[wave];
  float4 v = *(const float4*)(x + s * DIM + lane * 4);
  float* o = h + d * DIM + lane * 4;
  atomicAdd(o + 0, v.x * w);
  atomicAdd(o + 1, v.y * w);
  atomicAdd(o + 2, v.z * w);
  atomicAdd(o + 3, v.w * w);
}

// ---------------------------------------------------------------------------
// Kernel 3: fused  out = relu(h@W1+b1)@W2 + b2 [+ x@Wl + bl]
// All three 128x128 weight matrices live in LDS. Each wave computes a 16-row
// slab via V_WMMA_F32_16X16X4_F32; A fragments are register-resident (k-tile
// loops FULLY unrolled so the fragment arrays are never dynamically indexed).
// ---------------------------------------------------------------------------
__global__ void __launch_bounds__(256, 1)
gin_mlp_kernel(const float* __restrict__ h,
               const float* __restrict__ x,
               const float* __restrict__ W1g, const float* __restrict__ b1g,
               const float* __restrict__ W2g, const float* __restrict__ b2g,
               const float* __restrict__ Wlg, const float* __restrict__ blg,
               const int* __restrict__ flagp,
               float* __restrict__ out) {
  __shared__ float sW1[DIM * WSTRIDE];
  __shared__ float sW2[DIM * WSTRIDE];
  __shared__ float sWl[DIM * WSTRIDE];
  __shared__ float sT[WAVES_PER_BLOCK][16 * TSTRIDE];

  const int tid = threadIdx.x;

  // cooperative weight load global -> LDS (padded rows)
  for (int i = tid; i < DIM * DIM; i += blockDim.x) {
    int r = i >> 7, cc = i & 127;
    sW1[r * WSTRIDE + cc] = W1g[i];
    sW2[r * WSTRIDE + cc] = W2g[i];
    sWl[r * WSTRIDE + cc] = Wlg[i];
  }
  __syncthreads();

  const int wave = tid >> 5;
  const int lane = tid & 31;
  const int half = lane >> 4;   // 0: lanes 0-15, 1: lanes 16-31
  const int l16  = lane & 15;
  const int withLinear = *flagp;

  const int slab = blockIdx.x * WAVES_PER_BLOCK + wave;
  if (slab >= NSLABS) return;   // uniform per wave -> EXEC all-ones inside
  const long row0 = (long)slab * 16;

  float* myT = &sT[wave][0];

  // ---- A1 fragments: h slab (16 x 128), K-tile t covers K=[4t,4t+3] ----
  // lane layout per ISA: v0 = K=4t+2*half, v1 = K=4t+2*half+1, row = l16
  v2f a1[32];
  {
    const float* base = h + (row0 + l16) * DIM + 2 * half;
    #pragma unroll
    for (int t = 0; t < 32; ++t) a1[t] = *(const v2f*)(base + 4 * t);
  }

  // ---- layer 1: stage relu(h@W1 + b1) into wave-private LDS tile ----
  #pragma unroll 1
  for (int c = 0; c < 8; ++c) {
    const int n0 = c * 16;
    v8f acc = {0.f, 0.f, 0.f, 0.f, 0.f, 0.f, 0.f, 0.f};
    #pragma unroll
    for (int t = 0; t < 32; ++t) {
      const int k = 4 * t + 2 * half;   // B rows: k (v0), k+1 (v1)
      v2f b;
      b.x = sW1[k * WSTRIDE + n0 + l16];
      b.y = sW1[(k + 1) * WSTRIDE + n0 + l16];
      acc = __builtin_amdgcn_wmma_f32_16x16x4_f32(
          false, a1[t], false, b, (short)0, acc, false, false);
    }
    const float bias = b1g[n0 + l16];
    #pragma unroll
    for (int r = 0; r < 8; ++r) {       // D: VGPR r -> rows r, r+8
      float v = acc[r] + bias;
      v = v > 0.f ? v : 0.f;
      myT[(r + 8 * half) * TSTRIDE + n0 + l16] = v;
    }
  }
  // same-wave LDS staging: DScnt ordering handled by compiler, no barrier.

  // ---- A3 fragments: x slab for the residual GEMM (reuses a1's registers) ----
  v2f a3[32];
  if (withLinear) {
    const float* base = x + (row0 + l16) * DIM + 2 * half;
    #pragma unroll
    for (int t = 0; t < 32; ++t) a3[t] = *(const v2f*)(base + 4 * t);
  }

  // ---- layer 2 (+ residual) ----
  #pragma unroll 1
  for (int c = 0; c < 8; ++c) {
    const int n0 = c * 16;
    v8f acc = {0.f, 0.f, 0.f, 0.f, 0.f, 0.f, 0.f, 0.f};
    #pragma unroll
    for (int t = 0; t < 32; ++t) {
      const int k = 4 * t + 2 * half;
      v2f a2 = *(const v2f*)(myT + l16 * TSTRIDE + k);  // 8B-aligned LDS load
      v2f b;
      b.x = sW2[k * WSTRIDE + n0 + l16];
      b.y = sW2[(k + 1) * WSTRIDE + n0 + l16];
      acc = __builtin_amdgcn_wmma_f32_16x16x4_f32(
          false, a2, false, b, (short)0, acc, false, false);
    }
    if (withLinear) {
      #pragma unroll
      for (int t = 0; t < 32; ++t) {
        const int k = 4 * t + 2 * half;
        v2f b;
        b.x = sWl[k * WSTRIDE + n0 + l16];
        b.y = sWl[(k + 1) * WSTRIDE + n0 + l16];
        acc = __builtin_amdgcn_wmma_f32_16x16x4_f32(
            false, a3[t], false, b, (short)0, acc, false, false);
      }
    }
    const float bias = b2g[n0 + l16] + (withLinear ? blg[n0 + l16] : 0.f);
    float* obase = out + row0 * DIM + n0 + l16;
    #pragma unroll
    for (int r = 0; r < 8; ++r)
      obase[(long)(r + 8 * half) * DIM] = acc[r] + bias;
  }
}

// ---------------------------------------------------------------------------
extern "C" void kernel_launch(void* const* d_in, const int* in_sizes, int n_in,
                              void* d_out, int out_size, void* d_ws, size_t ws_size,
                              hipStream_t stream) {
  const float*     x    = (const float*)d_in[0];
  const long long* ei   = (const long long*)d_in[1];
  const float*     ew   = (const float*)d_in[2];
  /* d_in[3] = edge_attr (unused by reference) */
  const int*       flag = (const int*)d_in[4];
  const float*     W1   = (const float*)d_in[5];
  const float*     b1   = (const float*)d_in[6];
  const float*     W2   = (const float*)d_in[7];
  const float*     b2   = (const float*)d_in[8];
  const float*     Wl   = (const float*)d_in[9];
  const float*     bl   = (const float*)d_in[10];
  float* out = (float*)d_out;
  float* h   = (float*)d_ws;   // N_NODES*DIM floats = 51.2 MB scratch

  const int n4 = N_NODES * DIM / 4;
  init_h_kernel<<<(n4 + 255) / 256, 256, 0, stream>>>(x, h, n4);

  const long scatter_threads = (long)N_EDGES * 32;
  scatter_kernel<<<(int)((scatter_threads + 255) / 256), 256, 0, stream>>>(x, ei, ew, h);

  gin_mlp_kernel<<<(NSLABS + WAVES_PER_BLOCK - 1) / WAVES_PER_BLOCK, 256, 0, stream>>>(
      h, x, W1, b1, W2, b2, Wl, bl, flag, out);
}